// Distiller_6107443495111
// MI455X (gfx1250) — compile-verified
//
#include <hip/hip_runtime.h>

// Problem constants (fixed by the reference).
#define NROWS   8192
#define C_SEEN  5000
#define TOTAL_C 20000

#define COLS_PER_BLK 1024   // 256 threads x 4 columns
#define ROWS_PER_BLK 64

typedef unsigned int u32x4 __attribute__((ext_vector_type(4)));
typedef int          i32x8 __attribute__((ext_vector_type(8)));
typedef int          i32x4 __attribute__((ext_vector_type(4)));
typedef float        f32x4 __attribute__((ext_vector_type(4)));

// ---- Pass 1a: inv[c] = -1 for all 20000 classes ----------------------------
__global__ void fill_inv_kernel(int* __restrict__ inv) {
    int i = blockIdx.x * blockDim.x + threadIdx.x;
    if (i < TOTAL_C) inv[i] = -1;
}

// ---- Pass 1b: inv[seen[i]] = i  (seen ids are unique & in-range) -----------
__global__ void build_inv_kernel(const int* __restrict__ seen, int* __restrict__ inv) {
    int i = blockIdx.x * blockDim.x + threadIdx.x;
    if (i < C_SEEN) {
        int c = seen[i];
        if (c >= 0 && c < TOTAL_C) inv[c] = i;
    }
}

// ---- Pass 2: fused zero-fill + column gather for both matrices -------------
// Each block: 1024 output columns x 64 rows. inv[] slice staged into LDS via
// the Tensor Data Mover (TENSOR_LOAD_TO_LDS), waited with s_wait_tensorcnt.
__global__ __launch_bounds__(256)
void gather_scatter_kernel(const float* __restrict__ in_s,
                           const float* __restrict__ in_t,
                           const int*   __restrict__ inv,
                           float* __restrict__ out_s,
                           float* __restrict__ out_t) {
    __shared__ i32x4 sh_idx[COLS_PER_BLK / 4];   // 4 KB of column indices

    const int tid    = threadIdx.x;
    const int c0_blk = blockIdx.x * COLS_PER_BLK;
    const int row0   = blockIdx.y * ROWS_PER_BLK;

    if (tid < 32) {
        // Wave 0 issues one TDM DMA: inv[c0_blk .. c0_blk+1023] -> LDS.
        // tensor_dim0 is clamped to the remaining columns so OOB reads
        // return zero instead of touching memory past the buffer.
        unsigned long long gaddr = (unsigned long long)(size_t)(inv + c0_blk);
        unsigned int ldsaddr = (unsigned int)(size_t)(&sh_idx[0]); // flat addr low 32 = LDS offset
        unsigned int dim0 = (unsigned int)(TOTAL_C - c0_blk);

        u32x4 g0;
        g0[0] = 1u;                                   // count=1, user descriptor, no gather
        g0[1] = ldsaddr;                              // lds_addr (bytes)
        g0[2] = (unsigned int)gaddr;                  // global_addr[31:0]
        g0[3] = (unsigned int)((gaddr >> 32) & 0x01FFFFFFu) | (2u << 30); // addr[56:32] | type=2

        i32x8 g1;
        g1[0] = (int)(2u << 16);                      // data_size = 4 bytes; no multicast/pad
        g1[1] = (int)((dim0 & 0xFFFFu) << 16);        // tensor_dim0[15:0]
        g1[2] = (int)(((dim0 >> 16) & 0xFFFFu) | (1u << 16)); // tensor_dim0[31:16], tensor_dim1=1
        g1[3] = (int)((unsigned int)COLS_PER_BLK << 16);      // tile_dim0 = 1024
        g1[4] = 1;                                    // tile_dim1 = 1, tile_dim2 = 0
        g1[5] = (int)dim0;                            // tensor_dim0_stride (1 row -> don't care)
        g1[6] = 0;
        g1[7] = 0;

        i32x4 gz4 = {0, 0, 0, 0};                     // groups 2/3 unused (<=2D)
        i32x8 gz8 = {0, 0, 0, 0, 0, 0, 0, 0};         // extra group (6-arg toolchain form)
        __builtin_amdgcn_tensor_load_to_lds(g0, g1, gz4, gz4, gz8, 0);
        __builtin_amdgcn_s_wait_tensorcnt(0);
    }
    __syncthreads();

    const int c = c0_blk + tid * 4;
    const i32x4 idx = sh_idx[tid];                    // ds_load_b128

    for (int r = 0; r < ROWS_PER_BLK; ++r) {
        const int b = row0 + r;
        const float* __restrict__ rs = in_s + (size_t)b * C_SEEN;
        const float* __restrict__ rt = in_t + (size_t)b * C_SEEN;

        f32x4 vs, vt;
        vs.x = (idx.x >= 0) ? rs[idx.x] : 0.0f;
        vs.y = (idx.y >= 0) ? rs[idx.y] : 0.0f;
        vs.z = (idx.z >= 0) ? rs[idx.z] : 0.0f;
        vs.w = (idx.w >= 0) ? rs[idx.w] : 0.0f;
        vt.x = (idx.x >= 0) ? rt[idx.x] : 0.0f;
        vt.y = (idx.y >= 0) ? rt[idx.y] : 0.0f;
        vt.z = (idx.z >= 0) ? rt[idx.z] : 0.0f;
        vt.w = (idx.w >= 0) ? rt[idx.w] : 0.0f;

        const size_t ob = (size_t)b * TOTAL_C + c;
        if (c + 3 < TOTAL_C) {
            // Fully coalesced 16B non-temporal streaming stores.
            __builtin_nontemporal_store(vs, (f32x4*)(out_s + ob));
            __builtin_nontemporal_store(vt, (f32x4*)(out_t + ob));
        } else {
#pragma unroll
            for (int j = 0; j < 4; ++j) {
                if (c + j < TOTAL_C) {
                    out_s[ob + j] = vs[j];
                    out_t[ob + j] = vt[j];
                }
            }
        }
    }
}

extern "C" void kernel_launch(void* const* d_in, const int* in_sizes, int n_in,
                              void* d_out, int out_size, void* d_ws, size_t ws_size,
                              hipStream_t stream) {
    const float* in_s = (const float*)d_in[0];
    const float* in_t = (const float*)d_in[1];
    const int*   seen = (const int*)d_in[2];     // int32 on device (JAX default x64-off)
    (void)in_sizes; (void)n_in; (void)out_size; (void)ws_size;

    int* inv = (int*)d_ws;                       // 20000 * 4 B = 80 KB scratch

    float* out_s = (float*)d_out;
    float* out_t = (float*)d_out + (size_t)NROWS * TOTAL_C;

    fill_inv_kernel<<<(TOTAL_C + 255) / 256, 256, 0, stream>>>(inv);
    build_inv_kernel<<<(C_SEEN + 255) / 256, 256, 0, stream>>>(seen, inv);

    dim3 grid((TOTAL_C + COLS_PER_BLK - 1) / COLS_PER_BLK,   // 20
              NROWS / ROWS_PER_BLK);                          // 128
    gather_scatter_kernel<<<grid, 256, 0, stream>>>(in_s, in_t, inv, out_s, out_t);
}